// TernaryQuantizer_41704132444220
// MI455X (gfx1250) — compile-verified
//
#include <hip/hip_runtime.h>
#include <cstdint>

// ---------------------------------------------------------------------------
// TernaryQuantizer for MI455X (gfx1250).
//
// Math: product ternary codebooks factorize per-dim =>
//   q(x) = (e+ - e-) / (e0 + e+ + e-),  e_v = exp(-(x-v)^2 / T),  v in {-1,0,+1}
//   out  = x + BLEND * (q(x) - x),      BLEND = 1/(1+T), T = 0.3
// (gate cancels since q_factored == q_full exactly under the factorization)
//
// Memory-bound (6.3 MB total @ 23.3 TB/s ~ 0.27 us). Data path: CDNA5 async
// global->LDS (ASYNCcnt) double-buffered per wave; ds_load_b128 to VGPRs;
// global_store_b128 out.
// ---------------------------------------------------------------------------

__device__ __forceinline__ float soft_tern(float x) {
  constexpr float K2    = 4.80898346962987804f;   // log2(e) / 0.3
  constexpr float BLEND = 0.769230769230769231f;  // 1 / 1.3
  float d0 = x * x;            // dist^2 to 0
  float a  = x - 1.0f;
  float b  = x + 1.0f;
  float dp = a * a;            // dist^2 to +1
  float dm = b * b;            // dist^2 to -1
  float mn = fminf(d0, fminf(dp, dm));            // softmax max-shift (exact)
  float w0 = __builtin_amdgcn_exp2f((mn - d0) * K2);
  float wp = __builtin_amdgcn_exp2f((mn - dp) * K2);
  float wm = __builtin_amdgcn_exp2f((mn - dm) * K2);
  float q  = (wp - wm) * __builtin_amdgcn_rcpf(w0 + wp + wm);  // denom >= 1
  return fmaf(BLEND, q - x, x);
}

// Issue CDNA5 async copy: 16 B per lane, global (saddr + vaddr byte offset) -> LDS.
__device__ __forceinline__ void async_load_b128(uint32_t lds_byte, uint32_t gbl_byte,
                                                const float* base) {
  asm volatile("global_load_async_to_lds_b128 %0, %1, %2"
               :
               : "v"(lds_byte), "v"(gbl_byte), "s"(base)
               : "memory");
}
__device__ __forceinline__ void wait_async_le1() {
  asm volatile("s_wait_asynccnt 0x1" ::: "memory");
}
__device__ __forceinline__ void wait_async_le0() {
  asm volatile("s_wait_asynccnt 0x0" ::: "memory");
}

__global__ void __launch_bounds__(256)
ternary_vq_kernel(const float* __restrict__ x, float* __restrict__ out, int n) {
  constexpr int CHUNK = 128;                       // floats per wave-chunk (32 lanes x float4)
  __shared__ __attribute__((aligned(16))) float lds[8][2][CHUNK];  // 8 waves x 2 bufs = 8 KB

  const int lane = threadIdx.x & 31;
  const int wv   = threadIdx.x >> 5;
  const int wavesPerBlk = blockDim.x >> 5;
  const int gw  = blockIdx.x * wavesPerBlk + wv;   // global wave id
  const int nW  = gridDim.x * wavesPerBlk;         // total waves
  const int nChunks = n / CHUNK;

  // LDS byte offsets of this lane's 16B slot (low 32 bits of flat shared addr
  // == workgroup-relative LDS offset, which is what VDST of the async op wants).
  float* slot0 = &lds[wv][0][lane * 4];
  const uint32_t l0 = (uint32_t)(uintptr_t)slot0;
  const uint32_t l1 = l0 + (uint32_t)(CHUNK * sizeof(float));   // +512 B

  // Prologue: prefetch chunks gw and gw+nW (wave-uniform branches).
  if (gw < nChunks)
    async_load_b128(l0, (uint32_t)(gw * CHUNK + lane * 4) * 4u, x);
  if (gw + nW < nChunks)
    async_load_b128(l1, (uint32_t)((gw + nW) * CHUNK + lane * 4) * 4u, x);

  int it = 0;
  for (int c = gw; c < nChunks; c += nW, ++it) {
    // Async loads complete in order: <=1 outstanding means oldest (chunk c) landed.
    if (c + nW < nChunks) wait_async_le1();
    else                  wait_async_le0();

    const float4 v = *(const float4*)&lds[wv][it & 1][lane * 4];  // ds_load_b128
    float4 r;
    r.x = soft_tern(v.x);
    r.y = soft_tern(v.y);
    r.z = soft_tern(v.z);
    r.w = soft_tern(v.w);
    *(float4*)(out + c * CHUNK + lane * 4) = r;                   // global_store_b128

    // Refill the buffer we just consumed (ds_load already waited-on via use of v).
    int cn = c + 2 * nW;
    if (cn < nChunks)
      async_load_b128((it & 1) ? l1 : l0,
                      (uint32_t)(cn * CHUNK + lane * 4) * 4u, x);
  }

  // Scalar tail for n % CHUNK (empty for the reference sizes).
  for (int i = nChunks * CHUNK + (int)(blockIdx.x * blockDim.x + threadIdx.x);
       i < n; i += (int)(gridDim.x * blockDim.x))
    out[i] = soft_tern(x[i]);
}

extern "C" void kernel_launch(void* const* d_in, const int* in_sizes, int n_in,
                              void* d_out, int out_size, void* d_ws, size_t ws_size,
                              hipStream_t stream) {
  (void)n_in; (void)d_ws; (void)ws_size;
  const float* x = (const float*)d_in[0];   // (16, 8192, 6) f32
  // d_in[1] codebook, d_in[2] trigrams, d_in[3] gate_logit: mathematically
  // redundant (product-codebook factorization makes both branches identical,
  // so the sigmoid gate cancels) — intentionally unused.
  float* out = (float*)d_out;
  const int n = in_sizes[0];                // == out_size

  const int block = 256;                    // 8 wave32s
  int nChunks = n / 128;
  int blocks = (nChunks + 7) / 8;           // >= 1 chunk per wave
  if (blocks > 512) blocks = 512;           // cap; waves then stride 2 chunks
  if (blocks < 1)   blocks = 1;
  ternary_vq_kernel<<<blocks, block, 0, stream>>>(x, out, n);
}